// EdgeGNNLayer_90769838833733
// MI455X (gfx1250) — compile-verified
//
#include <hip/hip_runtime.h>

#define N_NODES 50000
#define N_EDGES 600000
#define E_TOT   (N_EDGES + N_NODES)
#define D       64
#define H       4
#define HD      256   // H*D
#define NEG_SLOPE 0.2f

typedef __attribute__((ext_vector_type(2))) float v2f;
typedef __attribute__((ext_vector_type(8))) float v8f;

// Monotonic float<->uint encoding so float segment-max can use atomicMax(u32).
__device__ __forceinline__ unsigned fenc(float f) {
  unsigned u = __float_as_uint(f);
  return (u & 0x80000000u) ? ~u : (u | 0x80000000u);
}
__device__ __forceinline__ float fdec(unsigned u) {
  return (u & 0x80000000u) ? __uint_as_float(u & 0x7FFFFFFFu)
                           : __uint_as_float(~u);
}
__device__ __forceinline__ float lrelu(float x) { return x > 0.f ? x : NEG_SLOPE * x; }

__device__ __forceinline__ void edge_sd(int e, const int* __restrict__ src,
                                        const int* __restrict__ dst, int& s, int& d) {
  if (e < N_EDGES) { s = src[e]; d = dst[e]; } else { s = e - N_EDGES; d = s; }
}

// -------- init: out[n,d] = sum_h gat_bias[h,d] (+ h_bias[d] on last layer);
//          menc = enc(-inf) sentinel (0), denom = 0
__global__ __launch_bounds__(256)
void gat_init(const float* __restrict__ gb, const float* __restrict__ hbias,
              float* __restrict__ out, unsigned* __restrict__ menc,
              float* __restrict__ denom, int add_hbias) {
  int idx = blockIdx.x * blockDim.x + threadIdx.x;
  if (idx >= N_NODES * D) return;
  int d = idx & (D - 1);
  float b = gb[d] + gb[D + d] + gb[2 * D + d] + gb[3 * D + d];
  if (add_hbias) b += hbias[d];
  out[idx] = b;
  if (d < H) { int n = idx >> 6; menc[n * H + d] = 0u; denom[n * H + d] = 0.f; }
}

// -------- fs = h @ W using V_WMMA_F32_16X16X4_F32 (exact f32 path).
// One wave per 16x16 output tile; K=64 -> 16 wmma issues.
__global__ __launch_bounds__(256)
void gat_gemm_fs(const float* __restrict__ h, const float* __restrict__ W,
                 float* __restrict__ fs) {
  const int wave = (blockIdx.x * blockDim.x + threadIdx.x) >> 5;  // tile id
  const int lane = threadIdx.x & 31;
  const int ct = wave & 15;          // col tile (HD/16 = 16)
  const int rt = wave >> 4;          // row tile (N/16 = 3125)
  const int m0 = rt * 16;
  const int n0 = ct * 16;
  const int half = lane >> 4;        // 0: lanes 0-15, 1: lanes 16-31
  const int l16  = lane & 15;

  const int arow = m0 + l16;         // A: lane holds row arow, K pair (half*2, half*2+1)
  const int bcol = n0 + l16;         // B: lane holds col bcol, same K pair
  v8f c = {};
  #pragma unroll
  for (int k0 = 0; k0 < D; k0 += 4) {
    const int ka = k0 + half * 2;
    v2f a, b;
    a.x = h[arow * D + ka];
    a.y = h[arow * D + ka + 1];
    b.x = W[ka * HD + bcol];
    b.y = W[(ka + 1) * HD + bcol];
    c = __builtin_amdgcn_wmma_f32_16x16x4_f32(false, a, false, b, (short)0, c,
                                              false, false);
  }
  // C/D layout: VGPR r -> row m0 + half*8 + r, col n0 + l16
  #pragma unroll
  for (int r = 0; r < 8; ++r)
    fs[(m0 + half * 8 + r) * HD + n0 + l16] = c[r];
}

// -------- per (node, head) attention scores
__global__ __launch_bounds__(256)
void gat_scores(const float* __restrict__ fs, const float* __restrict__ al,
                const float* __restrict__ ar, float* __restrict__ el,
                float* __restrict__ er) {
  int idx = blockIdx.x * blockDim.x + threadIdx.x;  // n*H + h
  if (idx >= N_NODES * H) return;
  int n = idx >> 2, hh = idx & 3;
  const float* f = fs + n * HD + hh * D;
  const float* a = al + hh * D;
  const float* b = ar + hh * D;
  float sl = 0.f, sr = 0.f;
  #pragma unroll
  for (int d = 0; d < D; ++d) { float v = f[d]; sl = fmaf(v, a[d], sl); sr = fmaf(v, b[d], sr); }
  el[idx] = sl; er[idx] = sr;
}

// -------- segment max over dst (encoded-uint atomicMax)
__global__ __launch_bounds__(256)
void gat_edge_max(const int* __restrict__ src, const int* __restrict__ dst,
                  const float* __restrict__ el, const float* __restrict__ er,
                  unsigned* __restrict__ menc) {
  int e = blockIdx.x * blockDim.x + threadIdx.x;
  if (e >= E_TOT) return;
  int s, d; edge_sd(e, src, dst, s, d);
  #pragma unroll
  for (int hh = 0; hh < H; ++hh) {
    float v = lrelu(el[s * H + hh] + er[d * H + hh]);
    atomicMax(&menc[d * H + hh], fenc(v));
  }
}

// -------- segment sum of exp(e - m[dst])
__global__ __launch_bounds__(256)
void gat_edge_sum(const int* __restrict__ src, const int* __restrict__ dst,
                  const float* __restrict__ el, const float* __restrict__ er,
                  const unsigned* __restrict__ menc, float* __restrict__ denom) {
  int e = blockIdx.x * blockDim.x + threadIdx.x;
  if (e >= E_TOT) return;
  int s, d; edge_sd(e, src, dst, s, d);
  #pragma unroll
  for (int hh = 0; hh < H; ++hh) {
    float v = lrelu(el[s * H + hh] + er[d * H + hh]);
    atomicAdd(&denom[d * H + hh], expf(v - fdec(menc[d * H + hh])));
  }
}

// -------- per-edge softmax weights
__global__ __launch_bounds__(256)
void gat_edge_alpha(const int* __restrict__ src, const int* __restrict__ dst,
                    const float* __restrict__ el, const float* __restrict__ er,
                    const unsigned* __restrict__ menc, const float* __restrict__ denom,
                    float* __restrict__ alpha) {
  int e = blockIdx.x * blockDim.x + threadIdx.x;
  if (e >= E_TOT) return;
  int s, d; edge_sd(e, src, dst, s, d);
  #pragma unroll
  for (int hh = 0; hh < H; ++hh) {
    float v = lrelu(el[s * H + hh] + er[d * H + hh]);
    float ex = expf(v - fdec(menc[d * H + hh]));
    alpha[e * H + hh] = ex / denom[d * H + hh];
  }
}

// -------- aggregate: out[dst,d] += sum_h alpha[e,h] * fs[src,h,d]
// 64 threads per edge (one per d); heads fused -> 64 atomics/edge not 256.
__global__ __launch_bounds__(256)
void gat_edge_aggr(const int* __restrict__ src, const int* __restrict__ dst,
                   const float* __restrict__ fs, const float* __restrict__ alpha,
                   float* __restrict__ out) {
  int idx = blockIdx.x * blockDim.x + threadIdx.x;
  int e = idx >> 6;
  if (e >= E_TOT) return;
  int d = idx & (D - 1);
  int s, dd; edge_sd(e, src, dst, s, dd);
  const float* f = fs + s * HD + d;
  const float* a = alpha + e * H;
  float v = fmaf(a[0], f[0],
           fmaf(a[1], f[D],
           fmaf(a[2], f[2 * D], a[3] * f[3 * D])));
  atomicAdd(&out[dd * D + d], v);
}

extern "C" void kernel_launch(void* const* d_in, const int* in_sizes, int n_in,
                              void* d_out, int out_size, void* d_ws, size_t ws_size,
                              hipStream_t stream) {
  const float* feat     = (const float*)d_in[0];
  const int*   src      = (const int*)d_in[1];
  const int*   dst      = (const int*)d_in[2];
  const float* Ws       = (const float*)d_in[3];
  const float* attn_l   = (const float*)d_in[4];
  const float* attn_r   = (const float*)d_in[5];
  const float* gat_bias = (const float*)d_in[6];
  const float* h_bias   = (const float*)d_in[7];
  float* out = (float*)d_out;

  char* ws = (char*)d_ws;
  float*    fs    = (float*)ws;    ws += (size_t)N_NODES * HD * sizeof(float);
  float*    el    = (float*)ws;    ws += (size_t)N_NODES * H * sizeof(float);
  float*    er    = (float*)ws;    ws += (size_t)N_NODES * H * sizeof(float);
  unsigned* menc  = (unsigned*)ws; ws += (size_t)N_NODES * H * sizeof(unsigned);
  float*    denom = (float*)ws;    ws += (size_t)N_NODES * H * sizeof(float);
  float*    alpha = (float*)ws;    ws += (size_t)E_TOT * H * sizeof(float);
  float*    hA    = (float*)ws;    ws += (size_t)N_NODES * D * sizeof(float);
  float*    hB    = (float*)ws;

  const float* hin[4]  = { feat, hA, hB, hA };
  float*       hout[4] = { hA, hB, hA, out };

  const int eb = (E_TOT + 255) / 256;
  for (int l = 0; l < 4; ++l) {
    const float* W  = Ws       + (size_t)l * D * HD;
    const float* al = attn_l   + (size_t)l * H * D;
    const float* ar = attn_r   + (size_t)l * H * D;
    const float* gb = gat_bias + (size_t)l * H * D;

    gat_init<<<(N_NODES * D + 255) / 256, 256, 0, stream>>>(gb, h_bias, hout[l],
                                                            menc, denom, l == 3);
    // 50000 tiles, 8 waves (tiles) per 256-thread block -> 6250 blocks
    gat_gemm_fs<<<(N_NODES / 16) * (HD / 16) / 8, 256, 0, stream>>>(hin[l], W, fs);
    gat_scores<<<(N_NODES * H + 255) / 256, 256, 0, stream>>>(fs, al, ar, el, er);
    gat_edge_max<<<eb, 256, 0, stream>>>(src, dst, el, er, menc);
    gat_edge_sum<<<eb, 256, 0, stream>>>(src, dst, el, er, menc, denom);
    gat_edge_alpha<<<eb, 256, 0, stream>>>(src, dst, el, er, menc, denom, alpha);
    gat_edge_aggr<<<(E_TOT * 64 + 255) / 256, 256, 0, stream>>>(src, dst, fs, alpha, hout[l]);
  }
}